// ModularQTS_NLP_16544214024234
// MI455X (gfx1250) — compile-verified
//
#include <hip/hip_runtime.h>
#include <hip/hip_bf16.h>

// ---------------------------------------------------------------------------
// ModularQTS NLP forward for gfx1250 (MI455X).
//  K0: positional-encoding table -> ws
//  K1: embed-gather + PE + mask + projection GEMM via V_WMMA_F32_16X16X4_F32,
//      then sigmoid*2pi -> angle table in ws
//  K2: 256 independent quantum statevector sims (one WG per (chunk,batch)),
//      2048-amplitude state resident in LDS, per-gate parallel pair updates
//  K3: chunk-mean + head projection -> d_out (32x2 f32)
// ---------------------------------------------------------------------------

typedef float v2f __attribute__((ext_vector_type(2)));
typedef float v8f __attribute__((ext_vector_type(8)));

#define NQ     8
#define NA     3
#define TOTAL  11
#define NSTATE 2048     // 2^11
#define CHUNK  8
#define NCHUNKS 8
#define LSEQ   64
#define NROTS  32
#define EMBED  512
#define BATCH  32
#define NOBS   7
#define TWO_PI 6.283185307179586476925f

// ---------------- K0: positional encoding table --------------------------
__global__ void pe_kernel(float* __restrict__ pe) {
  int idx = blockIdx.x * 256 + threadIdx.x;
  if (idx >= LSEQ * EMBED) return;
  int l = idx >> 9;          // / 512
  int c = idx & 511;
  int k2 = (c >> 1) << 1;    // even index into arange(0,512,2)*...
  float d = __expf(-(float)k2 * (logf(10000.0f) / (float)EMBED));
  float arg = (float)l * d;
  pe[idx] = (c & 1) ? cosf(arg) : sinf(arg);
}

// ---------------- K1: projection GEMM with f32 WMMA ----------------------
// C[m,n] = sum_k A[m,k] * W_proj[n,k],   m in [0,2048), n in [0,32), k in [0,512)
// A[m,k] = mask[m] * ( (id==0 ? 0 : emb[id,k]) + PE[l,k] )
// One wave computes one 16x16 tile; 128 K-steps of v_wmma_f32_16x16x4_f32.
__global__ __launch_bounds__(128) void proj_kernel(
    const float* __restrict__ emb, const float* __restrict__ Wp,
    const float* __restrict__ bp, const int* __restrict__ ids,
    const int* __restrict__ amask, const float* __restrict__ pe,
    float* __restrict__ TS) {
  int lane = threadIdx.x & 31;
  int wid  = threadIdx.x >> 5;
  int gw   = blockIdx.x * 4 + wid;   // 0..255 wave-tiles
  int tm   = gw >> 1;                // 0..127
  int tn   = gw & 1;                 // 0..1
  int m0 = tm * 16, n0 = tn * 16;

  // A fragment mapping (32-bit A 16x4): lanes 0-15 hold K={0,1}, lanes 16-31 K={2,3}
  int mloc = lane & 15;
  int m    = m0 + mloc;
  int kb   = (lane >> 4) * 2;

  int l  = m & 63;
  int id = ids[m];
  float maskv  = (float)amask[m];
  float escale = (id == 0) ? 0.0f : 1.0f;   // padding_idx = 0 row zeroed
  const float* embrow = emb + (size_t)id * EMBED;
  const float* perow  = pe + (size_t)l * EMBED;

  // B fragment: column n = n0 + (lane&15); W_proj stored row-major [32][512]
  int n = n0 + mloc;
  const float* wrow = Wp + (size_t)n * EMBED;

  __builtin_prefetch(embrow, 0, 0);          // global_prefetch_b8
  __builtin_prefetch(wrow, 0, 0);

  v8f c = {};
  for (int k0 = 0; k0 < EMBED; k0 += 4) {
    int ka = k0 + kb;
    v2f a, b;
    a.x = (embrow[ka]     * escale + perow[ka])     * maskv;
    a.y = (embrow[ka + 1] * escale + perow[ka + 1]) * maskv;
    b.x = wrow[ka];
    b.y = wrow[ka + 1];
    c = __builtin_amdgcn_wmma_f32_16x16x4_f32(
        /*neg_a=*/false, a, /*neg_b=*/false, b,
        /*c_mod=*/(short)0, c, /*reuse_a=*/false, /*reuse_b=*/false);
  }

  // D layout: VGPR r -> row m0+r (lanes 0-15) / m0+8+r (lanes 16-31), col = n0+(lane&15)
  int mrow = m0 + ((lane >> 4) ? 8 : 0);
  int ncol = n0 + (lane & 15);
  float bias = bp[ncol];
  for (int r = 0; r < 8; ++r) {
    float v  = c[r] + bias;
    float sg = 1.0f / (1.0f + __expf(-v));
    TS[(mrow + r) * NROTS + ncol] = sg * TWO_PI;
  }
}

// ---------------- K2: quantum statevector simulation ---------------------
// wire w -> bit (10-w).  main qubits 0..7 -> bits 10..3, ancilla 0..2 -> bits 2..0.

struct SimOp { unsigned char kind, pidx, a, b; };  // kind 0=RY(a), 1=CRX(ctrl=a,tgt=b)
__device__ __constant__ SimOp SIMOPS[32] = {
  {0,0,0,0},{0,1,1,0},{0,2,2,0},{0,3,3,0},{0,4,4,0},{0,5,5,0},{0,6,6,0},{0,7,7,0},
  {1,8,7,0},{1,9,6,7},{1,10,5,6},{1,11,4,5},{1,12,3,4},{1,13,2,3},{1,14,1,2},{1,15,0,1},
  {0,16,0,0},{0,17,1,0},{0,18,2,0},{0,19,3,0},{0,20,4,0},{0,21,5,0},{0,22,6,0},{0,23,7,0},
  {1,24,7,6},{1,25,0,7},{1,26,1,0},{1,27,2,1},{1,28,3,2},{1,29,4,3},{1,30,5,4},{1,31,6,5}
};

__device__ __forceinline__ int expand1(int p, int b) {
  int lo = p & ((1 << b) - 1);
  return ((p >> b) << (b + 1)) | lo;
}

// RY gate on bit `b`, fixed (c,s); 1024 pairs / 256 threads
__device__ __forceinline__ void ry_bit(float* re, float* im, float c, float s,
                                       int b, int tid) {
  for (int p = tid; p < NSTATE / 2; p += 256) {
    int i0 = expand1(p, b), i1 = i0 | (1 << b);
    float a0r = re[i0], a0i = im[i0], a1r = re[i1], a1i = im[i1];
    re[i0] = c * a0r - s * a1r;  im[i0] = c * a0i - s * a1i;
    re[i1] = s * a0r + c * a1r;  im[i1] = s * a0i + c * a1i;
  }
  __syncthreads();
}

// RZ gate on bit `b`: amp *= exp(-i*t/2) for bit=0, exp(+i*t/2) for bit=1
__device__ __forceinline__ void rz_bit(float* re, float* im, float c, float s,
                                       int b, int tid) {
  for (int idx = tid; idx < NSTATE; idx += 256) {
    float s2 = ((idx >> b) & 1) ? s : -s;
    float ar = re[idx], ai = im[idx];
    re[idx] = c * ar - s2 * ai;
    im[idx] = c * ai + s2 * ar;
  }
  __syncthreads();
}

// CX: swap target-bit pair where ctrl bit == 1
__device__ __forceinline__ void cx_bits(float* re, float* im, int bc, int bt, int tid) {
  int blo = bc < bt ? bc : bt, bhi = bc < bt ? bt : bc;
  for (int q = tid; q < NSTATE / 4; q += 256) {
    int base = expand1(expand1(q, blo), bhi);
    int i0 = base | (1 << bc);
    int i1 = i0 | (1 << bt);
    float tr = re[i0], ti = im[i0];
    re[i0] = re[i1]; im[i0] = im[i1];
    re[i1] = tr;     im[i1] = ti;
  }
  __syncthreads();
}

// Projector-controlled phase: e^{+i phi} on anc==0 subspace, e^{-i phi} elsewhere
__device__ __forceinline__ void pcphase_dev(float* re, float* im, float cp, float sp,
                                            int tid) {
  for (int idx = tid; idx < NSTATE; idx += 256) {
    float ss = ((idx & 7) == 0) ? sp : -sp;
    float ar = re[idx], ai = im[idx];
    re[idx] = cp * ar - ss * ai;
    im[idx] = cp * ai + ss * ar;
  }
  __syncthreads();
}

// One sim14 layer (32 gates). perTok: angle tables indexed [t*32+pidx] where
// t = ancilla value (low 3 idx bits); else shared [pidx]. adj: reversed order,
// negated angles (cos even, sin odd -> s *= -1).
__device__ void sim14_dev(float* re, float* im, const float* ctab, const float* stab,
                          bool perTok, bool adj, int tid) {
  float sgn = adj ? -1.0f : 1.0f;
  for (int step = 0; step < 32; ++step) {
    SimOp op = SIMOPS[adj ? 31 - step : step];
    if (op.kind == 0) {                      // RY on main wire op.a
      int b = 10 - op.a;
      for (int p = tid; p < NSTATE / 2; p += 256) {
        int i0 = expand1(p, b), i1 = i0 | (1 << b);
        int t = i0 & 7;
        float c = perTok ? ctab[t * 32 + op.pidx] : ctab[op.pidx];
        float s = (perTok ? stab[t * 32 + op.pidx] : stab[op.pidx]) * sgn;
        float a0r = re[i0], a0i = im[i0], a1r = re[i1], a1i = im[i1];
        re[i0] = c * a0r - s * a1r;  im[i0] = c * a0i - s * a1i;
        re[i1] = s * a0r + c * a1r;  im[i1] = s * a0i + c * a1i;
      }
    } else {                                 // CRX ctrl=op.a tgt=op.b
      int bc = 10 - op.a, bt = 10 - op.b;
      int blo = bc < bt ? bc : bt, bhi = bc < bt ? bt : bc;
      for (int q = tid; q < NSTATE / 4; q += 256) {
        int base = expand1(expand1(q, blo), bhi);
        int i0 = base | (1 << bc);
        int i1 = i0 | (1 << bt);
        int t = i0 & 7;
        float c = perTok ? ctab[t * 32 + op.pidx] : ctab[op.pidx];
        float s = (perTok ? stab[t * 32 + op.pidx] : stab[op.pidx]) * sgn;
        float a0r = re[i0], a0i = im[i0], a1r = re[i1], a1i = im[i1];
        // RX: n0 = c*a0 - i*s*a1 ; n1 = -i*s*a0 + c*a1
        re[i0] = c * a0r + s * a1i;  im[i0] = c * a0i - s * a1r;
        re[i1] = c * a1r + s * a0i;  im[i1] = c * a1i - s * a0r;
      }
    }
    __syncthreads();
  }
}

// prepare / prepare^dagger on ancilla (anc q -> bit 2-q)
__device__ void prepare_dev(float* re, float* im,
                            const float* ryc, const float* rys,
                            const float* rzc, const float* rzs,
                            bool adj, int tid) {
  for (int lyi = 0; lyi < 3; ++lyi) {
    int ly = adj ? 2 - lyi : lyi;
    if (!adj) {
      for (int q = 0; q < 3; ++q) {
        int e = ly * 3 + q, b = 2 - q;
        ry_bit(re, im, ryc[e], rys[e], b, tid);
        rz_bit(re, im, rzc[e], rzs[e], b, tid);
      }
      cx_bits(re, im, 2, 1, tid);   // ctrl anc0 -> tgt anc1
      cx_bits(re, im, 1, 0, tid);   // ctrl anc1 -> tgt anc2
    } else {
      cx_bits(re, im, 1, 0, tid);
      cx_bits(re, im, 2, 1, tid);
      for (int q = 2; q >= 0; --q) {
        int e = ly * 3 + q, b = 2 - q;
        rz_bit(re, im, rzc[e], -rzs[e], b, tid);
        ry_bit(re, im, ryc[e], -rys[e], b, tid);
      }
    }
  }
}

__global__ __launch_bounds__(256) void qsim_kernel(
    const float* __restrict__ TS, const float* __restrict__ prep,
    const float* __restrict__ sig, const float* __restrict__ qff,
    const float* __restrict__ Ap, const float* __restrict__ Bp,
    const float* __restrict__ Dp, float* __restrict__ EXPS) {
  __shared__ float re[NSTATE], im[NSTATE];
  __shared__ float tokc[CHUNK * NROTS], toks[CHUNK * NROTS];
  __shared__ float qffc[NROTS], qffs[NROTS];
  __shared__ float ryc[9], rys[9], rzc[9], rzs[9];
  __shared__ float cph[4], sph[4];
  __shared__ float Hre[NOBS][16], Him[NOBS][16];
  __shared__ float red[256];

  int tid   = threadIdx.x;
  int blk   = blockIdx.x;
  int chunk = blk >> 5;   // 0..7
  int batch = blk & 31;   // 0..31

  // ---- tables ----
  {
    int t = tid >> 5, p = tid & 31;   // 8 tokens x 32 params == 256 entries
    float ang = TS[(batch * LSEQ + chunk * CHUNK + t) * NROTS + p];
    tokc[tid] = cosf(0.5f * ang);
    toks[tid] = sinf(0.5f * ang);
  }
  if (tid < NROTS) {
    float a = qff[tid];
    qffc[tid] = cosf(0.5f * a); qffs[tid] = sinf(0.5f * a);
  }
  if (tid < 9) {
    float a = prep[tid * 2 + 0], z = prep[tid * 2 + 1];
    ryc[tid] = cosf(0.5f * a); rys[tid] = sinf(0.5f * a);
    rzc[tid] = cosf(0.5f * z); rzs[tid] = sinf(0.5f * z);
  }
  if (tid < 4) { cph[tid] = cosf(sig[tid]); sph[tid] = sinf(sig[tid]); }
  if (tid < NOBS) {
    const int RR[6] = {1, 2, 2, 3, 3, 3}, CC[6] = {0, 0, 1, 0, 1, 2};
    float hr[16], hi[16];
    for (int e = 0; e < 16; ++e) { hr[e] = 0.0f; hi[e] = 0.0f; }
    for (int e = 0; e < 6; ++e) {
      int r = RR[e], c = CC[e];
      float ar = Ap[tid * 6 + e], br = Bp[tid * 6 + e];
      hr[r * 4 + c] = ar;  hi[r * 4 + c] = br;    // lower = A + iB
      hr[c * 4 + r] = ar;  hi[c * 4 + r] = -br;   // upper = conj
    }
    for (int d = 0; d < 3; ++d) hr[d * 4 + d] = 2.0f * Dp[tid * 4 + 1 + d];
    for (int e = 0; e < 16; ++e) { Hre[tid][e] = hr[e]; Him[tid][e] = hi[e]; }
  }
  // |0...0>
  for (int idx = tid; idx < NSTATE; idx += 256) {
    re[idx] = (idx == 0) ? 1.0f : 0.0f;
    im[idx] = 0.0f;
  }
  __syncthreads();

  // ---- QSVT-style circuit ----
  pcphase_dev(re, im, cph[0], sph[0], tid);
  for (int k = 0; k < 3; ++k) {
    prepare_dev(re, im, ryc, rys, rzc, rzs, /*adj=*/false, tid);
    sim14_dev(re, im, tokc, toks, /*perTok=*/true, /*adj=*/(k & 1) != 0, tid);
    prepare_dev(re, im, ryc, rys, rzc, rzs, /*adj=*/true, tid);
    pcphase_dev(re, im, cph[k + 1], sph[k + 1], tid);
  }
  sim14_dev(re, im, qffc, qffs, /*perTok=*/false, /*adj=*/false, tid);

  // ---- expectation values: wires (w, w+1) -> bits (10-w, 9-w) ----
  for (int w = 0; w < NOBS; ++w) {
    int sh = 9 - w;
    float acc = 0.0f;
    for (int idx = tid; idx < NSTATE; idx += 256) {
      int i = (idx >> sh) & 3;
      int base = idx & ~(3 << sh);
      float sir = re[idx], sii = im[idx];
      float tr = 0.0f, ti = 0.0f;
      for (int j = 0; j < 4; ++j) {
        int sib = base | (j << sh);
        float hjr = Hre[w][j * 4 + i], hji = Him[w][j * 4 + i];
        float sjr = re[sib], sji = im[sib];
        tr += hjr * sjr + hji * sji;   // Re(H[j,i]*conj(s_j))
        ti += hji * sjr - hjr * sji;   // Im(H[j,i]*conj(s_j))
      }
      acc += sir * tr - sii * ti;      // Re(s_i * H[j,i]*conj(s_j))
    }
    red[tid] = acc;
    __syncthreads();
    for (int off = 128; off > 0; off >>= 1) {
      if (tid < off) red[tid] += red[tid + off];
      __syncthreads();
    }
    if (tid == 0) EXPS[(chunk * BATCH + batch) * NOBS + w] = red[0];
    __syncthreads();
  }
}

// ---------------- K3: chunk-mean + head ----------------------------------
__global__ void head_kernel(const float* __restrict__ EXPS,
                            const float* __restrict__ Wh,
                            const float* __restrict__ bh,
                            float* __restrict__ out) {
  int tid = threadIdx.x;
  if (tid >= BATCH * 2) return;
  int b = tid >> 1, o = tid & 1;
  float acc = bh[o];
  for (int w = 0; w < NOBS; ++w) {
    float mean = 0.0f;
    for (int c = 0; c < NCHUNKS; ++c) mean += EXPS[(c * BATCH + b) * NOBS + w];
    acc += (mean * 0.125f) * Wh[o * NOBS + w];
  }
  out[b * 2 + o] = acc;
}

// ---------------- launch ---------------------------------------------------
extern "C" void kernel_launch(void* const* d_in, const int* in_sizes, int n_in,
                              void* d_out, int out_size, void* d_ws, size_t ws_size,
                              hipStream_t stream) {
  const float* emb   = (const float*)d_in[0];
  const float* Wp    = (const float*)d_in[1];
  const float* bp    = (const float*)d_in[2];
  const float* prep  = (const float*)d_in[3];
  const float* sig   = (const float*)d_in[4];
  const float* qff   = (const float*)d_in[5];
  const float* Ap    = (const float*)d_in[6];
  const float* Bp    = (const float*)d_in[7];
  const float* Dp    = (const float*)d_in[8];
  const float* Wh    = (const float*)d_in[9];
  const float* bh    = (const float*)d_in[10];
  const int*   ids   = (const int*)d_in[11];
  const int*   amask = (const int*)d_in[12];
  float* out = (float*)d_out;

  char* ws = (char*)d_ws;
  float* PEt  = (float*)(ws);                                      // 64*512 f32
  float* TS   = (float*)(ws + (size_t)LSEQ * EMBED * 4);           // 2048*32 f32
  float* EXPS = (float*)(ws + (size_t)LSEQ * EMBED * 4
                            + (size_t)BATCH * LSEQ * NROTS * 4);   // 256*7 f32

  pe_kernel<<<(LSEQ * EMBED + 255) / 256, 256, 0, stream>>>(PEt);
  proj_kernel<<<64, 128, 0, stream>>>(emb, Wp, bp, ids, amask, PEt, TS);
  qsim_kernel<<<NCHUNKS * BATCH, 256, 0, stream>>>(TS, prep, sig, qff, Ap, Bp, Dp, EXPS);
  head_kernel<<<1, 64, 0, stream>>>(EXPS, Wh, bh, out);
}